// BypassModel_53815940219080
// MI455X (gfx1250) — compile-verified
//
#include <hip/hip_runtime.h>
#include <math.h>
#include <stdint.h>

// Problem constants (match reference)
#define B_   64
#define S_   2048
#define D_   1024
#define T_   16
#define NS_  (B_ * T_)      // 1024 segments / rows
#define EPS_ 1e-5f

typedef __attribute__((ext_vector_type(2))) float v2f;
typedef __attribute__((ext_vector_type(4))) float v4f;
typedef __attribute__((ext_vector_type(8))) float v8f;

// ---------------------------------------------------------------------------
// Kernel 1: ragged segment max + mean pooling over hidden_states.
// One block per (batch, turn) segment; 256 threads x float4 = 1024 cols/row.
// Streams 512 MB of HBM -> pure bandwidth; dominant cost of the whole net.
// ---------------------------------------------------------------------------
__global__ __launch_bounds__(256)
void pool_kernel(const float* __restrict__ h,
                 const int* __restrict__ turns,
                 const int* __restrict__ parts,
                 float* __restrict__ pooled) {
    const int seg = blockIdx.x;          // 0..NS-1, ordered (i, j) like reference
    const int i = seg / T_;
    const int j = seg % T_;

    // segment [start, end): ends = 1 + cumsum(parts); T=16 so scalar scan is free
    int start = 1, e = 1;
    for (int jj = 0; jj <= j; ++jj) {
        start = e;
        e += parts[i * T_ + jj];
    }
    int end = e;
    if (j >= turns[i]) { start = 1; end = 1; }   // inactive turn -> empty segment
    if (end > S_) end = S_;

    const int c0 = threadIdx.x * 4;              // 4 contiguous features per lane
    const float* base = h + (size_t)i * S_ * D_;

    float4 mx = make_float4(-INFINITY, -INFINITY, -INFINITY, -INFINITY);
    float4 sm = make_float4(0.f, 0.f, 0.f, 0.f);

    for (int p = start; p < end; ++p) {
        if (p + 2 < end)
            __builtin_prefetch(base + (size_t)(p + 2) * D_ + c0, 0, 0);
        const float4 v = *(const float4*)(base + (size_t)p * D_ + c0);
        mx.x = fmaxf(mx.x, v.x); mx.y = fmaxf(mx.y, v.y);
        mx.z = fmaxf(mx.z, v.z); mx.w = fmaxf(mx.w, v.w);
        sm.x += v.x; sm.y += v.y; sm.z += v.z; sm.w += v.w;
    }

    const int cnt = end - start;
    const float inv = (cnt > 0) ? (1.0f / (float)cnt) : 0.0f;
    if (cnt == 0) mx = make_float4(0.f, 0.f, 0.f, 0.f);

    float* pm = pooled + (size_t)seg * (2 * D_);
    *(float4*)(pm + c0) = mx;                                        // cols [0, D)   = max
    float4 mean = make_float4(sm.x * inv, sm.y * inv, sm.z * inv, sm.w * inv);
    *(float4*)(pm + D_ + c0) = mean;                                 // cols [D, 2D)  = mean
}

// ---------------------------------------------------------------------------
// Kernel 2: fp32 WMMA GEMM  C[M,N] = A[M,K] @ W[K,N] + bias, optional tanh.
// Block: 256 threads = 8 waves (wave32). Block tile 128x128, K-chunk 16.
// Double-buffered LDS staged with GLOBAL_LOAD_ASYNC_TO_LDS_B128 (ASYNCcnt),
// compute via V_WMMA_F32_16X16X4_F32 (wave tile 64x32 = 4x2 v8f accumulators).
// ---------------------------------------------------------------------------
#define BLK_M 128
#define BLK_N 128
#define KC    16

__global__ __launch_bounds__(256)
void gemm_wmma(const float* __restrict__ A, const float* __restrict__ W,
               const float* __restrict__ bias, float* __restrict__ C,
               int M, int N, int K, int act) {
    __shared__ __align__(16) float As[2][BLK_M][KC];   // m-major: float2 A-frags
    __shared__ __align__(16) float Bs[2][KC][BLK_N];   // k-major: row-striped B

    const int tid  = threadIdx.x;
    const int lane = tid & 31;         // wave32 lane
    const int wave = tid >> 5;         // 0..7
    const int wr   = wave >> 2;        // 0..1  (wave row: 64 rows each)
    const int wc   = wave & 3;         // 0..3  (wave col: 32 cols each)
    const int rowBase = blockIdx.y * BLK_M;
    const int colBase = blockIdx.x * BLK_N;

    v8f acc[4][2] = {};                // 4 M-tiles x 2 N-tiles of 16x16

    // cooperative tile-copy indices (2 x B128 per thread per tile)
    const int arow = tid >> 1;              // 0..127
    const int acol = (tid & 1) * 8;         // 0 or 8
    const int bkk  = tid >> 4;              // 0..15
    const int bcol = (tid & 15) * 8;        // 0..120

    // Issue the 4 async B128 copies for one K-chunk into LDS buffer `buf`.
    auto issue_chunk = [&](int kc, int buf) {
        unsigned long long ga =
            (unsigned long long)(A + (size_t)(rowBase + arow) * K + kc + acol);
        unsigned la = (unsigned)(uintptr_t)&As[buf][arow][acol];
        asm volatile("global_load_async_to_lds_b128 %0, %1, off"
                     :: "v"(la), "v"(ga) : "memory");
        asm volatile("global_load_async_to_lds_b128 %0, %1, off"
                     :: "v"(la + 16u), "v"(ga + 16ull) : "memory");
        unsigned long long gb =
            (unsigned long long)(W + (size_t)(kc + bkk) * N + colBase + bcol);
        unsigned lb = (unsigned)(uintptr_t)&Bs[buf][bkk][bcol];
        asm volatile("global_load_async_to_lds_b128 %0, %1, off"
                     :: "v"(lb), "v"(gb) : "memory");
        asm volatile("global_load_async_to_lds_b128 %0, %1, off"
                     :: "v"(lb + 16u), "v"(gb + 16ull) : "memory");
    };

    const int nch = K / KC;
    issue_chunk(0, 0);

    for (int c = 0; c < nch; ++c) {
        const int buf = c & 1;
        if (c + 1 < nch) {
            issue_chunk((c + 1) * KC, buf ^ 1);
            // 4 newest (chunk c+1) may remain outstanding; chunk c has landed
            asm volatile("s_wait_asynccnt 0x4" ::: "memory");
        } else {
            asm volatile("s_wait_asynccnt 0x0" ::: "memory");
        }
        __syncthreads();    // chunk c visible to all waves

#pragma unroll
        for (int ks = 0; ks < KC; ks += 4) {
            // ISA A 16x4 layout: lane = (k>=2)*16 + m, vgpr = k&1 -> float2/lane
            const int kk = ks + 2 * (lane >> 4);
            v2f af[4];
            v2f bf[2];
#pragma unroll
            for (int mt = 0; mt < 4; ++mt) {
                const int ml = wr * 64 + mt * 16 + (lane & 15);
                af[mt] = *(const v2f*)&As[buf][ml][kk];
            }
#pragma unroll
            for (int nt = 0; nt < 2; ++nt) {
                const int nl = wc * 32 + nt * 16 + (lane & 15);
                v2f bv;
                bv[0] = Bs[buf][kk][nl];
                bv[1] = Bs[buf][kk + 1][nl];
                bf[nt] = bv;
            }
#pragma unroll
            for (int mt = 0; mt < 4; ++mt)
#pragma unroll
                for (int nt = 0; nt < 2; ++nt)
                    acc[mt][nt] = __builtin_amdgcn_wmma_f32_16x16x4_f32(
                        /*neg_a=*/false, af[mt], /*neg_b=*/false, bf[nt],
                        /*c_mod=*/(short)0, acc[mt][nt],
                        /*reuse_a=*/false, /*reuse_b=*/false);
        }
        __syncthreads();    // all reads of buf done before next iter overwrites it
    }

    // Epilogue. C/D layout: VGPR r -> row r (lanes 0-15) / row r+8 (lanes 16-31),
    // col = lane & 15 within the 16x16 tile.
#pragma unroll
    for (int mt = 0; mt < 4; ++mt) {
#pragma unroll
        for (int nt = 0; nt < 2; ++nt) {
            const int col = colBase + wc * 32 + nt * 16 + (lane & 15);
            const float bb = bias ? bias[col] : 0.0f;
#pragma unroll
            for (int r = 0; r < 8; ++r) {
                const int row = rowBase + wr * 64 + mt * 16 + r + 8 * (lane >> 4);
                float v = acc[mt][nt][r] + bb;
                if (act) v = tanhf(v);
                C[(size_t)row * N + col] = v;
            }
        }
    }
}

// ---------------------------------------------------------------------------
// Kernel 3: BatchNorm1d training-mode column stats (biased var) ->
// fold into affine: a = g * rsqrt(var + eps); c = beta - mean * a.
// ---------------------------------------------------------------------------
__global__ __launch_bounds__(256)
void bn_stats(const float* __restrict__ Y, const float* __restrict__ g,
              const float* __restrict__ beta, float* __restrict__ bnp,
              int rows, int cols) {
    const int c = blockIdx.x * blockDim.x + threadIdx.x;
    if (c >= cols) return;
    float s = 0.f, s2 = 0.f;
    for (int r = 0; r < rows; ++r) {
        const float v = Y[(size_t)r * cols + c];
        s += v; s2 += v * v;
    }
    const float invn = 1.0f / (float)rows;
    const float m = s * invn;
    const float var = s2 * invn - m * m;
    const float a = g[c] * rsqrtf(var + EPS_);
    bnp[c] = a;
    bnp[cols + c] = beta[c] - m * a;
}

// Kernel 4: fused BN-affine + tanh elementwise
__global__ __launch_bounds__(256)
void bn_apply(const float* __restrict__ Y, const float* __restrict__ bnp,
              float* __restrict__ X, int cols, int total) {
    const int idx = blockIdx.x * blockDim.x + threadIdx.x;
    if (idx >= total) return;
    const int c = idx % cols;
    X[idx] = tanhf(Y[idx] * bnp[c] + bnp[cols + c]);
}

// ---------------------------------------------------------------------------
// Kernel 5: tiny head GEMM [1024,1024] @ [1024,nout] + bias (nout <= 102).
// One block per output row; X row cached in LDS; negligible FLOPs.
// ---------------------------------------------------------------------------
__global__ __launch_bounds__(128)
void head_gemm(const float* __restrict__ X, const float* __restrict__ Wf,
               const float* __restrict__ bf, float* __restrict__ out, int nout) {
    __shared__ float Xs[D_];
    const int m = blockIdx.x;
    for (int k = threadIdx.x; k < D_; k += blockDim.x)
        Xs[k] = X[(size_t)m * D_ + k];
    __syncthreads();
    for (int n = threadIdx.x; n < nout; n += blockDim.x) {
        float acc = bf[n];
        for (int k = 0; k < D_; ++k)
            acc += Xs[k] * Wf[(size_t)k * nout + n];
        out[(size_t)m * nout + n] = acc;
    }
}

// ---------------------------------------------------------------------------
extern "C" void kernel_launch(void* const* d_in, const int* in_sizes, int n_in,
                              void* d_out, int out_size, void* d_ws, size_t ws_size,
                              hipStream_t stream) {
    const float* hidden    = (const float*)d_in[0];
    const int*   turns     = (const int*)d_in[1];
    const int*   parts     = (const int*)d_in[2];
    const float* pooler_W  = (const float*)d_in[3];
    const float* pooler_b  = (const float*)d_in[4];
    const float* pool_W    = (const float*)d_in[5];
    const float* pool_b    = (const float*)d_in[6];
    const float* pool_g    = (const float*)d_in[7];
    const float* pool_beta = (const float*)d_in[8];

    // workspace layout (floats): pooled | X | Y | X0 | bn params
    float* ws     = (float*)d_ws;
    float* pooled = ws;                               // NS x 2D   (8 MB)
    float* X      = pooled + (size_t)NS_ * 2 * D_;    // NS x D    (4 MB)
    float* Y      = X + (size_t)NS_ * D_;             // NS x D    (4 MB)
    float* X0     = Y + (size_t)NS_ * D_;             // NS x D    (4 MB)
    float* bnp    = X0 + (size_t)NS_ * D_;            // 2 x D

    const dim3 gemmGrid(D_ / BLK_N, NS_ / BLK_M);     // (8, 8)
    const int bnApplyBlocks = (NS_ * D_) / 256;

    // 1) ragged pooling (bandwidth phase: 512 MB stream)
    pool_kernel<<<NS_, 256, 0, stream>>>(hidden, turns, parts, pooled);

    // 2) pooler: tanh(pooled @ W[2D,D] + b)  -> X
    gemm_wmma<<<gemmGrid, 256, 0, stream>>>(pooled, pooler_W, pooler_b, X,
                                            NS_, D_, 2 * D_, /*act=*/1);

    // 3) pool Linear_Block: Linear -> BN -> tanh  -> X0 (shared by 3 towers)
    gemm_wmma<<<gemmGrid, 256, 0, stream>>>(X, pool_W, pool_b, Y, NS_, D_, D_, 0);
    bn_stats<<<D_ / 256, 256, 0, stream>>>(Y, pool_g, pool_beta, bnp, NS_, D_);
    bn_apply<<<bnApplyBlocks, 256, 0, stream>>>(Y, bnp, X0, D_, NS_ * D_);

    // 4) three towers (L2-resident: ~60 MB weights << 192 MB L2)
    const int nouts[3] = {7, 5, 102};
    size_t outOff = 0;
    for (int t = 0; t < 3; ++t) {
        const float* Ws    = (const float*)d_in[9  + 6 * t];
        const float* bs    = (const float*)d_in[10 + 6 * t];
        const float* gs    = (const float*)d_in[11 + 6 * t];
        const float* betas = (const float*)d_in[12 + 6 * t];
        const float* Wf    = (const float*)d_in[13 + 6 * t];
        const float* bf    = (const float*)d_in[14 + 6 * t];

        const float* cur = X0;
        for (int k = 0; k < 4; ++k) {
            gemm_wmma<<<gemmGrid, 256, 0, stream>>>(cur, Ws + (size_t)k * D_ * D_,
                                                    bs + (size_t)k * D_, Y,
                                                    NS_, D_, D_, 0);
            bn_stats<<<D_ / 256, 256, 0, stream>>>(Y, gs + (size_t)k * D_,
                                                   betas + (size_t)k * D_, bnp, NS_, D_);
            bn_apply<<<bnApplyBlocks, 256, 0, stream>>>(Y, bnp, X, D_, NS_ * D_);
            cur = X;
        }
        head_gemm<<<NS_, 128, 0, stream>>>(cur, Wf, bf, (float*)d_out + outOff, nouts[t]);
        outOff += (size_t)NS_ * nouts[t];
    }
}